// Vista3DLoss_30734785970765
// MI455X (gfx1250) — compile-verified
//
#include <hip/hip_runtime.h>
#include <math.h>

#define NV   1048576      // 64*128*128 voxels per batch
#define BD   2
#define NC   8
#define ED   16
#define KSEG 17           // NUM_INST + 1
#define DZv  64
#define DYv  128
#define DXv  128

// ---- workspace layout: 2 doubles, then floats, then optional w-flag bytes ----
#define WSF_INTER 0        // [2][8]
#define WSF_CARDP 16       // [2][8]
#define WSF_CARDG 32       // [2][8]
#define WSF_WSUM  48       // [2][17]
#define WSF_CNT   82       // [2][17]
#define WSF_CNUM  116      // [2][17][16]
#define WSF_CEN   660      // [2][17][16]
#define WSF_CSQ   1204     // [2][17]
#define WSF_PULL  1238     // [2][17]
#define WSF_PUSH  1272     // [2]
#define WSF_NPRES 1274     // [2]
#define WSF_NORM  1276     // [2]
#define WSF_TOTAL 1278
#define WS_DBL_BYTES 16
#define WS_BYTES (WS_DBL_BYTES + WSF_TOTAL * 4)

typedef __attribute__((ext_vector_type(2))) float v2f;
typedef __attribute__((ext_vector_type(8))) float v8f;

__device__ __forceinline__ float rlanef(float v, int l) {
    return __int_as_float(__builtin_amdgcn_readlane(__float_as_int(v), l));
}

// 3x3x3 edge-clamped min/max boundary weight
__device__ __forceinline__ float bweight(const int* __restrict__ lab, int v) {
    int z = v >> 14, y = (v >> 7) & 127, x = v & 127;
    int mx = -2147483647, mn = 2147483647;
#pragma unroll
    for (int dz = -1; dz <= 1; ++dz) {
        int zz = min(max(z + dz, 0), DZv - 1);
#pragma unroll
        for (int dy = -1; dy <= 1; ++dy) {
            int yy = min(max(y + dy, 0), DYv - 1);
#pragma unroll
            for (int dx = -1; dx <= 1; ++dx) {
                int xx = min(max(x + dx, 0), DXv - 1);
                int t = lab[(zz << 14) | (yy << 7) | xx];
                mx = max(mx, t);
                mn = min(mn, t);
            }
        }
    }
    return (mx != mn) ? 10.0f : 1.0f;
}

__device__ __forceinline__ float wred(float v) {
    v += __shfl_xor(v, 1, 32);
    v += __shfl_xor(v, 2, 32);
    v += __shfl_xor(v, 4, 32);
    v += __shfl_xor(v, 8, 32);
    v += __shfl_xor(v, 16, 32);
    return v;
}

// ---- Pass A: CE + Dice partials. Register accumulators + wave reduction ----
__global__ void __launch_bounds__(256) k_sem(const float* __restrict__ logits,
                                             const int* __restrict__ cls,
                                             double* wsd, float* wsf) {
    __shared__ float s_acc[BD * 26];   // per batch: 8 inter, 8 cardp, 8 cardg, nll, valid
    int tid = threadIdx.x, lane = tid & 31;
    for (int i = tid; i < BD * 26; i += 256) s_acc[i] = 0.f;
    __syncthreads();

    int gid = blockIdx.x * 256 + tid;
    int stride = gridDim.x * 256;

    for (int b = 0; b < BD; ++b) {
        float inter[NC] = {0}, cardp[NC] = {0}, cardg[NC] = {0};
        float nll_s = 0.f, val_s = 0.f;
        const float* Lb = logits + (size_t)b * NC * NV;
        const int* Cb = cls + (size_t)b * NV;

        for (int v = gid; v < NV; v += stride) {
            int t = Cb[v];
            bool valid = (t != -100);
            int ts = valid ? min(max(t, 0), NC - 1) : 0;
            float xo[NC], mx = -INFINITY, lt = 0.f;
#pragma unroll
            for (int c = 0; c < NC; ++c) {
                xo[c] = Lb[(size_t)c * NV + v];
                mx = fmaxf(mx, xo[c]);
                lt = (c == ts) ? xo[c] : lt;
            }
            float s = 0.f, p[NC];
#pragma unroll
            for (int c = 0; c < NC; ++c) { p[c] = expf(xo[c] - mx); s += p[c]; }
            float inv = 1.f / s;
            if (valid) {
                nll_s += logf(s) - (lt - mx);
                val_s += 1.f;
#pragma unroll
                for (int c = 0; c < NC; ++c) {
                    float pc = p[c] * inv;
                    bool h = (c == ts);
                    cardp[c] += pc;
                    inter[c] += h ? pc : 0.f;
                    cardg[c] += h ? 1.f : 0.f;
                }
            }
        }
#pragma unroll
        for (int c = 0; c < NC; ++c) {
            float r0 = wred(inter[c]);
            float r1 = wred(cardp[c]);
            float r2 = wred(cardg[c]);
            if (lane == 0) {
                atomicAdd(&s_acc[b * 26 + c], r0);
                atomicAdd(&s_acc[b * 26 + 8 + c], r1);
                atomicAdd(&s_acc[b * 26 + 16 + c], r2);
            }
        }
        float rn = wred(nll_s), rv = wred(val_s);
        if (lane == 0) {
            atomicAdd(&s_acc[b * 26 + 24], rn);
            atomicAdd(&s_acc[b * 26 + 25], rv);
        }
    }
    __syncthreads();
    if (tid < BD * 26) {
        int b = tid / 26, q = tid % 26;
        float v = s_acc[tid];
        if (q < 8)        atomicAdd(&wsf[WSF_INTER + b * NC + q], v);
        else if (q < 16)  atomicAdd(&wsf[WSF_CARDP + b * NC + (q - 8)], v);
        else if (q < 24)  atomicAdd(&wsf[WSF_CARDG + b * NC + (q - 16)], v);
        else if (q == 24) atomicAdd(&wsd[0], (double)v);
        else              atomicAdd(&wsd[1], (double)v);
    }
}

// ---- Pass B: segment sums via WMMA, 32 voxels/iteration (2 subtiles),
//      readlane broadcasts, register-resident accumulators. ----
__global__ void __launch_bounds__(256) k_seg(const float* __restrict__ embed,
                                             const int* __restrict__ labels,
                                             float* wsf, unsigned char* wbuf) {
    __shared__ float s_w[KSEG], s_c[KSEG], s_e[16 * ED];
    int tid = threadIdx.x;
    int wave = tid >> 5, lane = tid & 31;
    int m = lane & 15, hi = lane >> 4;
    int wgid = blockIdx.x * 8 + wave;
    int nwaves = gridDim.x * 8;

    for (int b = 0; b < BD; ++b) {
        for (int i = tid; i < KSEG; i += 256) { s_w[i] = 0.f; s_c[i] = 0.f; }
        for (int i = tid; i < 16 * ED; i += 256) s_e[i] = 0.f;
        __syncthreads();

        const float* eb = embed + (size_t)b * ED * NV;
        const int* lab = labels + (size_t)b * NV;
        v8f acc0 = {0.f, 0.f, 0.f, 0.f, 0.f, 0.f, 0.f, 0.f};
        v8f acc1 = {0.f, 0.f, 0.f, 0.f, 0.f, 0.f, 0.f, 0.f};

        for (int t = wgid; t < NV / 32; t += nwaves) {
            int base = t * 32;
            int v = base + lane;                 // every lane owns one voxel
            int lval = min(max(lab[v], 0), 16);
            float wval = bweight(lab, v);
            if (wbuf) wbuf[(size_t)b * NV + v] = (wval > 1.5f) ? 1 : 0;
            atomicAdd(&s_w[lval], wval);
            atomicAdd(&s_c[lval], 1.f);

            const float* bp = eb + (size_t)m * NV + base;
#pragma unroll
            for (int j = 0; j < 4; ++j) {        // K chunks of 4 voxels, 2 subtiles
                int k0 = 4 * j + 2 * hi;
                // subtile 0: voxels base+0..15 (source lanes 0..15)
                int l0 = __builtin_amdgcn_readlane(lval, k0);
                int l1 = __builtin_amdgcn_readlane(lval, k0 + 1);
                float w0 = rlanef(wval, k0);
                float w1 = rlanef(wval, k0 + 1);
                v2f a;
                a.x = (l0 == m + 1) ? w0 : 0.f;
                a.y = (l1 == m + 1) ? w1 : 0.f;
                v2f bv; bv.x = bp[k0]; bv.y = bp[k0 + 1];
                acc0 = __builtin_amdgcn_wmma_f32_16x16x4_f32(
                    false, a, false, bv, (short)0, acc0, false, false);
                // subtile 1: voxels base+16..31 (source lanes 16..31)
                int l2 = __builtin_amdgcn_readlane(lval, 16 + k0);
                int l3 = __builtin_amdgcn_readlane(lval, 16 + k0 + 1);
                float w2 = rlanef(wval, 16 + k0);
                float w3 = rlanef(wval, 16 + k0 + 1);
                v2f a2;
                a2.x = (l2 == m + 1) ? w2 : 0.f;
                a2.y = (l3 == m + 1) ? w3 : 0.f;
                v2f bv2; bv2.x = bp[16 + k0]; bv2.y = bp[16 + k0 + 1];
                acc1 = __builtin_amdgcn_wmma_f32_16x16x4_f32(
                    false, a2, false, bv2, (short)0, acc1, false, false);
            }
        }
        v8f acc = acc0 + acc1;
        // D(M=seg-1, N=dim): lane holds dim=m, segs (j + 8*hi)+1
#pragma unroll
        for (int j = 0; j < 8; ++j)
            atomicAdd(&s_e[(j + 8 * hi) * ED + m], acc[j]);
        __syncthreads();

        for (int i = tid; i < KSEG; i += 256) {
            atomicAdd(&wsf[WSF_WSUM + b * KSEG + i], s_w[i]);
            atomicAdd(&wsf[WSF_CNT + b * KSEG + i], s_c[i]);
        }
        for (int i = tid; i < 16 * ED; i += 256)
            atomicAdd(&wsf[WSF_CNUM + (b * KSEG + 1) * ED + i], s_e[i]);
        __syncthreads();
    }
}

// ---- Pass C: centers, push, norm (tiny) ----
__global__ void k_centers(float* wsf) {
    __shared__ float cen[BD][KSEG][ED];
    __shared__ float csq[BD][KSEG];
    __shared__ float pres[BD][KSEG];
    int tid = threadIdx.x;
    if (tid < BD * KSEG) {
        int b = tid / KSEG, k = tid % KSEG;
        float wsum = wsf[WSF_WSUM + b * KSEG + k];
        float inv = 1.f / (wsum + 1e-8f);
        float sq = 0.f;
        for (int c = 0; c < ED; ++c) {
            float cc = wsf[WSF_CNUM + (b * KSEG + k) * ED + c] * inv;
            cen[b][k][c] = cc;
            sq += cc * cc;
            wsf[WSF_CEN + (b * KSEG + k) * ED + c] = cc;
        }
        csq[b][k] = sq;
        wsf[WSF_CSQ + b * KSEG + k] = sq;
        pres[b][k] = (wsf[WSF_CNT + b * KSEG + k] > 0.f) ? 1.f : 0.f;
    }
    __syncthreads();
    if (tid < BD) {
        int b = tid;
        float np = 0.f, nrm = 0.f;
        for (int k = 1; k < KSEG; ++k) { np += pres[b][k]; nrm += pres[b][k] * sqrtf(csq[b][k]); }
        wsf[WSF_NPRES + b] = np;
        wsf[WSF_NORM + b] = nrm;
        float push = 0.f;
        for (int i = 0; i < 16; ++i)
            for (int j = i + 1; j < 16; ++j)
                if (pres[b][i + 1] > 0.f && pres[b][j + 1] > 0.f) {
                    float d2 = 0.f;
                    for (int c = 0; c < ED; ++c) {
                        float df = cen[b][i + 1][c] - cen[b][j + 1][c];
                        d2 += df * df;
                    }
                    float r = fmaxf(3.0f - sqrtf(d2), 0.f);   // 2*DELTA_D
                    push += r * r;
                }
        wsf[WSF_PUSH + b] = push;
    }
}

// ---- Pass D: pull loss via WMMA, 32 voxels/iteration (2 subtiles) ----
__global__ void __launch_bounds__(256) k_pull(const float* __restrict__ embed,
                                              const int* __restrict__ labels,
                                              const float* __restrict__ wsf_in,
                                              float* wsf_out,
                                              const unsigned char* __restrict__ wbuf) {
    __shared__ float cenL[BD][16][ED];
    __shared__ float csqL[BD][KSEG];
    __shared__ float Dtile[8][512];       // per wave: 2 subtiles x 16x16
    __shared__ float pullAcc[BD * KSEG];
    int tid = threadIdx.x;
    for (int i = tid; i < BD * 16 * ED; i += 256) {
        int b = i / (16 * ED), r = i % (16 * ED), k = r / ED, c = r % ED;
        cenL[b][k][c] = wsf_in[WSF_CEN + (b * KSEG + (k + 1)) * ED + c];
    }
    for (int i = tid; i < BD * KSEG; i += 256) csqL[i / KSEG][i % KSEG] = wsf_in[WSF_CSQ + i];
    for (int i = tid; i < BD * KSEG; i += 256) pullAcc[i] = 0.f;
    __syncthreads();

    int wave = tid >> 5, lane = tid & 31;
    int m = lane & 15, hi = lane >> 4;
    int wgid = blockIdx.x * 8 + wave;
    int nwaves = gridDim.x * 8;
    float* Dw = &Dtile[wave][0];

    for (int b = 0; b < BD; ++b) {
        const float* eb = embed + (size_t)b * ED * NV;
        const int* lab = labels + (size_t)b * NV;
        v2f bf[4];
#pragma unroll
        for (int j = 0; j < 4; ++j) {
            int d0 = 4 * j + 2 * hi;
            bf[j].x = cenL[b][m][d0];
            bf[j].y = cenL[b][m][d0 + 1];
        }
        for (int t = wgid; t < NV / 32; t += nwaves) {
            int base = t * 32;
            int vox0 = base + m, vox1 = base + 16 + m;
            v8f acc0 = {0.f, 0.f, 0.f, 0.f, 0.f, 0.f, 0.f, 0.f};
            v8f acc1 = {0.f, 0.f, 0.f, 0.f, 0.f, 0.f, 0.f, 0.f};
            float nrm0 = 0.f, nrm1 = 0.f;
#pragma unroll
            for (int j = 0; j < 4; ++j) {
                int d0 = 4 * j + 2 * hi;
                v2f a0, a1;
                a0.x = eb[(size_t)d0 * NV + vox0];
                a0.y = eb[(size_t)(d0 + 1) * NV + vox0];
                a1.x = eb[(size_t)d0 * NV + vox1];
                a1.y = eb[(size_t)(d0 + 1) * NV + vox1];
                nrm0 += a0.x * a0.x + a0.y * a0.y;
                nrm1 += a1.x * a1.x + a1.y * a1.y;
                acc0 = __builtin_amdgcn_wmma_f32_16x16x4_f32(
                    false, a0, false, bf[j], (short)0, acc0, false, false);
                acc1 = __builtin_amdgcn_wmma_f32_16x16x4_f32(
                    false, a1, false, bf[j], (short)0, acc1, false, false);
            }
            nrm0 += __shfl_xor(nrm0, 16, 32);
            nrm1 += __shfl_xor(nrm1, 16, 32);
#pragma unroll
            for (int j = 0; j < 8; ++j) {
                Dw[m * 16 + (j + 8 * hi)] = acc0[j];
                Dw[256 + m * 16 + (j + 8 * hi)] = acc1[j];
            }
            __builtin_amdgcn_wave_barrier();
            // every lane selects for its own voxel: lane L -> voxel base+L,
            // subtile = hi; its reduced norm is already in-lane.
            {
                int vsel = base + lane;
                float nrm = hi ? nrm1 : nrm0;
                int l = lab[vsel]; l = min(max(l, 0), 16);
                if (l > 0) {
                    float dp = Dw[hi * 256 + (l - 1) * 16 + m];
                    float d2 = fmaxf(nrm + csqL[b][l] - 2.f * dp, 0.f);
                    float r = fmaxf(sqrtf(d2) - 0.5f, 0.f);   // DELTA_V
                    if (r > 0.f) {
                        float w = wbuf ? (wbuf[(size_t)b * NV + vsel] ? 10.f : 1.f)
                                       : bweight(lab, vsel);
                        atomicAdd(&pullAcc[b * KSEG + l], r * r * w);
                    }
                }
            }
            __builtin_amdgcn_wave_barrier();
        }
    }
    __syncthreads();
    for (int i = tid; i < BD * KSEG; i += 256)
        atomicAdd(&wsf_out[WSF_PULL + i], pullAcc[i]);
}

// ---- Pass E: finalize ----
__global__ void k_final(const double* wsd, const float* wsf, float* out) {
    if (threadIdx.x != 0 || blockIdx.x != 0) return;
    float loss_ce = (float)(wsd[0] / fmax(wsd[1], 1.0));
    float dsum = 0.f;
    for (int i = 0; i < BD * NC; ++i) {
        float inter = wsf[WSF_INTER + i], cp = wsf[WSF_CARDP + i], cg = wsf[WSF_CARDG + i];
        dsum += (2.f * inter + 1e-5f) / (cp + cg + 1e-5f);
    }
    float loss_dice = 1.f - dsum / (float)(BD * NC);
    float loss_sem = loss_ce + loss_dice;

    float lp = 0.f, lpu = 0.f, ln = 0.f, val = 0.f;
    for (int b = 0; b < BD; ++b) {
        for (int k = 1; k < KSEG; ++k) {
            float cnt = wsf[WSF_CNT + b * KSEG + k];
            if (cnt > 0.f) lp += wsf[WSF_PULL + b * KSEG + k] / fmaxf(cnt, 1.f);
        }
        float np = wsf[WSF_NPRES + b];
        float npairs = np * (np - 1.f) * 0.5f;
        if (npairs > 0.f) lpu += wsf[WSF_PUSH + b] / fmaxf(npairs, 1.f);
        if (np > 0.f) { ln += wsf[WSF_NORM + b] / fmaxf(np, 1.f); val += 1.f; }
    }
    float n = fmaxf(val, 1.f);
    float loss_ins = (1.0f * lp + 1.0f * lpu + 0.001f * ln) / n;
    out[0] = loss_sem + loss_ins;
    out[1] = loss_sem;
    out[2] = loss_ce;
    out[3] = loss_dice;
    out[4] = loss_ins;
}

extern "C" void kernel_launch(void* const* d_in, const int* in_sizes, int n_in,
                              void* d_out, int out_size, void* d_ws, size_t ws_size,
                              hipStream_t stream) {
    (void)in_sizes; (void)n_in; (void)out_size;
    const float* sem = (const float*)d_in[0];
    const float* emb = (const float*)d_in[1];
    const int* cls = (const int*)d_in[2];
    const int* lab = (const int*)d_in[3];
    float* out = (float*)d_out;
    double* wsd = (double*)d_ws;
    float* wsf = (float*)((char*)d_ws + WS_DBL_BYTES);

    bool cacheW = ws_size >= (size_t)WS_BYTES + (size_t)BD * NV;
    unsigned char* wbuf = cacheW ? (unsigned char*)d_ws + WS_BYTES : (unsigned char*)0;

    hipMemsetAsync(d_ws, 0, WS_BYTES, stream);
    k_sem<<<512, 256, 0, stream>>>(sem, cls, wsd, wsf);
    k_seg<<<512, 256, 0, stream>>>(emb, lab, wsf, wbuf);
    k_centers<<<1, 64, 0, stream>>>(wsf);
    k_pull<<<512, 256, 0, stream>>>(emb, lab, wsf, wsf, wbuf);
    k_final<<<1, 1, 0, stream>>>(wsd, wsf, out);
}